// Phi_r_85804856639623
// MI455X (gfx1250) — compile-verified
//
#include <hip/hip_runtime.h>

// ---------------------------------------------------------------------------
// Anisotropic SPDE quadratic form q[b] on MI455X (gfx1250).
// Memory-bound: 75.5 MB input @ 23.3 TB/s ~ 3.3 us floor; ~0.45 GFLOP fp32.
// CDNA5 paths: global_load_async_to_lds_b128/_b32 (ASYNCcnt) double-buffered
// tile pipeline, s_wait_asynccnt, v_wmma_f32_16x16x4_f32 wave reduction.
// ---------------------------------------------------------------------------

#define KAPPA2 0.1089f   /* 0.33^2 */
#define GAMMA_C 1.0f
#define BETA_C 25.0f

constexpr int NB = 8;
constexpr int NTP = 9;      // planes per batch (7 field + vx + vy)
constexpr int NT = 7;
constexpr int NY = 512;
constexpr int NX = 512;

constexpr int TX = 64;              // output tile width
constexpr int TY = 16;              // output tile height
constexpr int NXT = NX / TX;        // 8  tiles in x
constexpr int NYT = NY / TY;        // 32 tiles in y
constexpr int HROWS = TY + 2;       // 18 halo rows   (gy0-1 .. gy0+16)
constexpr int LSTRIDE = 72;         // halo cols 0..71 <-> gx = gx0-4 .. gx0+67
                                    // (16B rows; stencil reads cols 3..68)
constexpr int NTHREADS = 256;       // 8 waves (wave32)
constexpr int NMAIN = HROWS * 16;   // 288 aligned b128 chunks (cols 4..67)
constexpr int NHALO = HROWS * 2;    // 36  halo b128 chunks (cols 0..3, 68..71)
constexpr int NESLOT = HROWS * 66;  // 1188 edge-path b32 slots

// Issue async DMA global->LDS for one plane's halo tile.
// interior: fully in-range halo, pure b128 chunks, shift-only addressing.
// edge:     b32 per in-range slot only; OOB slots hold pre-zeroed Dirichlet 0s.
__device__ __forceinline__ void issue_plane(const float* __restrict__ plane,
                                            float* buf, int gy0, int gx0,
                                            int tid, bool interior) {
    if (interior) {
        // Main region: rows 0..17, 16 chunks of 4 floats, gx0 .. gx0+63.
#pragma unroll
        for (int it = 0; it < 2; ++it) {
            int k = tid + it * NTHREADS;
            if (k < NMAIN) {
                int row = k >> 4;              // 0..17
                int ch  = k & 15;              // 0..15
                int gy  = gy0 - 1 + row;
                unsigned lds_off = (unsigned)(unsigned long long)
                    &buf[row * LSTRIDE + 4 + ch * 4];
                const float* src = plane + (size_t)gy * NX + gx0 + ch * 4;
                asm volatile("global_load_async_to_lds_b128 %0, %1, off"
                             :: "v"(lds_off), "v"(src)
                             : "memory");
            }
        }
        // Halo columns: per row, one chunk at gx0-4 and one at gx0+64.
        if (tid < NHALO) {
            int row  = tid >> 1;               // 0..17
            int side = tid & 1;                // 0 = left, 1 = right
            int col  = side ? 68 : 0;          // LDS cols 68..71 / 0..3
            int gy   = gy0 - 1 + row;
            int gxc  = gx0 - 4 + col;          // gx0-4 or gx0+64 (16B aligned)
            unsigned lds_off = (unsigned)(unsigned long long)
                &buf[row * LSTRIDE + col];
            const float* src = plane + (size_t)gy * NX + gxc;
            asm volatile("global_load_async_to_lds_b128 %0, %1, off"
                         :: "v"(lds_off), "v"(src)
                         : "memory");
        }
    } else {
        for (int s = tid; s < NESLOT; s += NTHREADS) {
            int row = s / 66;                  // 0..17
            int c   = s - row * 66 + 3;        // cols 3..68 (what stencil reads)
            int gy  = gy0 - 1 + row;
            int gx  = gx0 - 4 + c;             // gx0-1 .. gx0+64
            if ((unsigned)gy < (unsigned)NY && (unsigned)gx < (unsigned)NX) {
                unsigned lds_off = (unsigned)(unsigned long long)
                    &buf[row * LSTRIDE + c];
                const float* src = plane + (size_t)gy * NX + gx;
                asm volatile("global_load_async_to_lds_b32 %0, %1, off"
                             :: "v"(lds_off), "v"(src)
                             : "memory");
            }
        }
    }
}

__global__ void __launch_bounds__(NTHREADS, 2)
spde_q_kernel(const float* __restrict__ state, float* __restrict__ q) {
    const int bx = blockIdx.x;          // 0..7
    const int by = blockIdx.y;          // 0..31
    const int b  = blockIdx.z;          // 0..7
    const int tid = threadIdx.x;

    const int tx = tid & (TX - 1);          // 0..63
    const int tyBase = (tid >> 6) << 2;     // 0,4,8,12 (4 rows per thread)
    const int gx0 = bx * TX;
    const int gy0 = by * TY;
    const int gx = gx0 + tx;

    const bool interior = (bx > 0) & (bx < NXT - 1) & (by > 0) & (by < NYT - 1);

    __shared__ float tile[2][HROWS * LSTRIDE];   // double-buffered halo tiles
    __shared__ float red[NTHREADS / 32];         // per-wave partials

    const float* base = state + (size_t)b * NTP * NY * NX;
    const float* vxp = base + (size_t)7 * NY * NX;
    const float* vyp = base + (size_t)8 * NY * NX;

    // Per-pixel operator constants (center-only, no halo):
    //   Ax = CK*c - (H11*(l+r) + H22*(u+d) + H12h*t3),
    //   CK = k^2 + 2*(H11+H22),  H12h = 0.5*BETA*vx*vy = 12.5*vx*vy.
    float H11[4], H22[4], H12h[4], CK[4], prevc2[4];
#pragma unroll
    for (int r = 0; r < 4; ++r) {
        int gy = gy0 + tyBase + r;
        float vx = vxp[(size_t)gy * NX + gx];
        float vy = vyp[(size_t)gy * NX + gx];
        H11[r]  = GAMMA_C + BETA_C * vx * vx;
        H22[r]  = GAMMA_C + BETA_C * vy * vy;
        H12h[r] = 0.5f * BETA_C * vx * vy;
        CK[r]   = KAPPA2 + 2.0f * (H11[r] + H22[r]);
        prevc2[r] = 0.0f;
    }

    // Edge blocks: zero both buffers ONCE. DMA never touches OOB slots, so
    // the Dirichlet zeros persist across all planes (no per-plane stores).
    if (!interior) {
        float4* zp = (float4*)&tile[0][0];
#pragma unroll
        for (int i = tid; i < 2 * HROWS * LSTRIDE / 4; i += NTHREADS)
            zp[i] = make_float4(0.0f, 0.0f, 0.0f, 0.0f);
        asm volatile("s_wait_dscnt 0" ::: "memory");
    }
    __syncthreads();   // zeros visible before any DMA write can land

    float acc = 0.0f;

    // Prime the pipeline with plane 0.
    issue_plane(base, tile[0], gy0, gx0, tid, interior);

    for (int t = 0; t < NT; ++t) {
        // Wait this wave's DMA; block-sync so every wave's slots are visible;
        // then prefetch next plane (overlaps with the stencil below).
        asm volatile("s_wait_asynccnt 0" ::: "memory");
        __syncthreads();
        if (t + 1 < NT)
            issue_plane(base + (size_t)(t + 1) * NY * NX,
                        tile[(t + 1) & 1], gy0, gx0, tid, interior);

        const float* buf = tile[t & 1];
        const int lx = tx + 4;               // center col in halo layout

        // Load the 6x3 neighborhood once; slide over the 4 output rows.
        float rm[6], rc[6], rp[6];
#pragma unroll
        for (int rr = 0; rr < 6; ++rr) {
            const float* p = &buf[(tyBase + rr) * LSTRIDE + (lx - 1)];
            rm[rr] = p[0];
            rc[rr] = p[1];
            rp[rr] = p[2];
        }

#pragma unroll
        for (int r = 0; r < 4; ++r) {
            float c  = rc[r + 1];
            float t1 = rm[r + 1] + rp[r + 1];            // left + right
            float t2 = rc[r] + rc[r + 2];                // up + down
            float t3 = (rp[r + 2] + rm[r]) - (rm[r + 2] + rp[r]); // 4*uxy
            float d  = H11[r] * t1;
            d = fmaf(H22[r], t2, d);
            d = fmaf(H12h[r], t3, d);
            float Ax = fmaf(CK[r], c, -d);
            float Mx = c + Ax;                           // DT = 1

            if (t == 0) {
                acc = fmaf(Ax, Ax, acc);                 // ||Ax_0||^2
                acc = fmaf(1.05f * c, c, acc);           // 1.05*||x_0||^2
            } else {
                acc = fmaf(Mx, Mx, acc);                 // nM[1..6]
                acc = fmaf(-prevc2[r], Mx, acc);         // -2<x_{t-1},Mx_t>
                if (t < NT - 1) acc = fmaf(c, c, acc);   // nx[1..5]
            }
            prevc2[r] = c + c;                           // 2*x_t for next t
        }
    }

    // ---- wave reduction via v_wmma_f32_16x16x4_f32 with B = ones ----
    // A (16x4 f32): lanes 0-15 {v0=K0,v1=K1}, lanes 16-31 {v0=K2,v1=K3}.
    // a = {acc, 0} => D[m][n] = acc[m] + acc[m+16]; sum 8 D VGPRs per lane,
    // add lane^16 partner => full 32-lane sum.
    float wavesum;
#if __has_builtin(__builtin_amdgcn_wmma_f32_16x16x4_f32)
    {
        typedef float v2f __attribute__((ext_vector_type(2)));
        typedef float v8f __attribute__((ext_vector_type(8)));
        v2f a; a.x = acc;  a.y = 0.0f;
        v2f bb; bb.x = 1.0f; bb.y = 1.0f;
        v8f cz = {};
        v8f d = __builtin_amdgcn_wmma_f32_16x16x4_f32(
            false, a, false, bb, (short)0, cz, false, false);
        float s = d[0] + d[1] + d[2] + d[3] + d[4] + d[5] + d[6] + d[7];
        wavesum = s + __shfl_xor(s, 16, 32);
    }
#else
    {
        wavesum = acc;
        for (int off = 16; off > 0; off >>= 1)
            wavesum += __shfl_xor(wavesum, off, 32);
    }
#endif

    const int lane = tid & 31;
    const int wid = tid >> 5;
    if (lane == 0) red[wid] = wavesum;
    __syncthreads();
    if (tid == 0) {
        float tot = 0.0f;
#pragma unroll
        for (int w = 0; w < NTHREADS / 32; ++w) tot += red[w];
        atomicAdd(&q[b], tot);
    }
}

__global__ void zero_out_kernel(float* __restrict__ q, int n) {
    int i = blockIdx.x * blockDim.x + threadIdx.x;
    if (i < n) q[i] = 0.0f;
}

extern "C" void kernel_launch(void* const* d_in, const int* in_sizes, int n_in,
                              void* d_out, int out_size, void* d_ws, size_t ws_size,
                              hipStream_t stream) {
    (void)in_sizes; (void)n_in; (void)d_ws; (void)ws_size;
    const float* state = (const float*)d_in[0];
    float* q = (float*)d_out;

    zero_out_kernel<<<1, 64, 0, stream>>>(q, out_size);

    dim3 grid(NXT, NYT, NB);   // (8, 32, 8) = 2048 blocks
    spde_q_kernel<<<grid, NTHREADS, 0, stream>>>(state, q);
}